// network_representation_module_GAT_57346403336486
// MI455X (gfx1250) — compile-verified
//
#include <hip/hip_runtime.h>
#include <math.h>

// ---------------------------------------------------------------------------
// GAT (2 layers) for MI455X / gfx1250.
//  N=50000 nodes, E=800000 edges, 128 feats = 4 heads x 32 dims.
//  GEMM via V_WMMA_F32_16X16X4_F32 (fp32 matrix pipe, exact precision).
//  Edge softmax + aggregation via f32 atomics (all node arrays are L2-resident
//  on MI455X's 192MB L2, so the scatter runs at L2 bandwidth).
// ---------------------------------------------------------------------------

#define GAT_N      50000
#define GAT_E      800000
#define GAT_F      128     // HEADS*HID
#define GAT_HEADS  4
#define GAT_HID    32
#define NEG_SLOPE  0.2f

typedef float v2f __attribute__((ext_vector_type(2)));
typedef float v8f __attribute__((ext_vector_type(8)));

// --------------------------- GEMM: H = X @ W -------------------------------
// X: [N,128] row-major, W: [128,128] row-major, H: [N,128].
// One wave computes one 16x16 tile of H with 32 chained 16x16x4 f32 WMMAs.
// Block = 256 threads = 8 waves; wave w owns N-tile w (8 tiles * 16 = 128 cols).
// Grid.x = N/16 = 3125 M-tiles (N divisible by 16 exactly).
__global__ __launch_bounds__(256)
void gat_gemm_wmma(const float* __restrict__ X, const float* __restrict__ W,
                   float* __restrict__ H) {
  const int wave = threadIdx.x >> 5;        // 0..7  -> N tile
  const int lane = threadIdx.x & 31;
  const int half = lane >> 4;               // 0: lanes 0-15, 1: lanes 16-31
  const int l15  = lane & 15;
  const int mrow = blockIdx.x * 16 + l15;   // A-fragment row for this lane
  const int col  = wave * 16 + l15;         // B/D column for this lane

  const float* __restrict__ xrow = X + (size_t)mrow * GAT_F;

  v8f acc = {};
#pragma unroll 8
  for (int k0 = 0; k0 < GAT_F; k0 += 4) {
    const int ka = k0 + 2 * half;           // K pair handled by this lane-half
    // A 16x4 fragment: VGPR0 = K {0|2}, VGPR1 = K {1|3}  (contiguous -> b64 load)
    v2f a;
    a.x = xrow[ka];
    a.y = xrow[ka + 1];
    // B 4x16 fragment: VGPR0 = row K {0|2}, VGPR1 = row K {1|3}
    v2f b;
    b.x = W[(size_t)ka * GAT_F + col];
    b.y = W[(size_t)(ka + 1) * GAT_F + col];
    acc = __builtin_amdgcn_wmma_f32_16x16x4_f32(
        /*neg_a=*/false, a, /*neg_b=*/false, b,
        /*c_mod=*/(short)0, acc, /*reuse_a=*/false, /*reuse_b=*/false);
  }

  // D layout: VGPR r -> M = r (lanes 0-15) / M = 8+r (lanes 16-31), N = lane&15
  const int m0 = blockIdx.x * 16 + 8 * half;
#pragma unroll
  for (int r = 0; r < 8; ++r) {
    H[(size_t)(m0 + r) * GAT_F + col] = acc[r];
  }
}

// ------------------- per-node attention coefficients -----------------------
// el[n,h] = sum_d H[n,h,d]*al[h,d]; er likewise.
__global__ __launch_bounds__(256)
void gat_attn_coef(const float* __restrict__ H, const float* __restrict__ al,
                   const float* __restrict__ ar, float* __restrict__ el,
                   float* __restrict__ er) {
  const int n = blockIdx.x * blockDim.x + threadIdx.x;
  if (n >= GAT_N) return;
  const float* __restrict__ h = H + (size_t)n * GAT_F;
#pragma unroll
  for (int hd = 0; hd < GAT_HEADS; ++hd) {
    float sl = 0.f, sr = 0.f;
#pragma unroll
    for (int d = 0; d < GAT_HID; ++d) {
      const float v = h[hd * GAT_HID + d];
      sl = fmaf(v, al[hd * GAT_HID + d], sl);
      sr = fmaf(v, ar[hd * GAT_HID + d], sr);
    }
    el[n * GAT_HEADS + hd] = sl;
    er[n * GAT_HEADS + hd] = sr;
  }
}

// ----------------------------- init helpers --------------------------------
__global__ __launch_bounds__(256)
void gat_init_nodes(float* __restrict__ mx, float* __restrict__ den, int nh) {
  const int i = blockIdx.x * blockDim.x + threadIdx.x;
  if (i < nh) { mx[i] = -3.0e38f; den[i] = 0.f; }
}

__global__ __launch_bounds__(256)
void gat_zero(float* __restrict__ p, int n) {
  const int i = blockIdx.x * blockDim.x + threadIdx.x;
  if (i < n) p[i] = 0.f;
}

// --------------- edge logits: leakyrelu(el[src]+er[dst]), seg-max ----------
__global__ __launch_bounds__(256)
void gat_edge_logits(const float* __restrict__ el, const float* __restrict__ er,
                     const int* __restrict__ src, const int* __restrict__ dst,
                     float* __restrict__ eatt, float* __restrict__ mx) {
  const int e = blockIdx.x * blockDim.x + threadIdx.x;
  if (e >= GAT_E) return;
  const int s = src[e], d = dst[e];
#pragma unroll
  for (int h = 0; h < GAT_HEADS; ++h) {
    float v = el[s * GAT_HEADS + h] + er[d * GAT_HEADS + h];
    v = (v > 0.f) ? v : NEG_SLOPE * v;
    eatt[(size_t)e * GAT_HEADS + h] = v;
    atomicMax(&mx[d * GAT_HEADS + h], v);   // global_atomic_max_num_f32
  }
}

// --------------- edge exp + segment-sum denominator ------------------------
__global__ __launch_bounds__(256)
void gat_edge_exp(float* __restrict__ eatt, const int* __restrict__ dst,
                  const float* __restrict__ mx, float* __restrict__ den) {
  const int e = blockIdx.x * blockDim.x + threadIdx.x;
  if (e >= GAT_E) return;
  const int d = dst[e];
#pragma unroll
  for (int h = 0; h < GAT_HEADS; ++h) {
    const float a = __expf(eatt[(size_t)e * GAT_HEADS + h] - mx[d * GAT_HEADS + h]);
    eatt[(size_t)e * GAT_HEADS + h] = a;
    atomicAdd(&den[d * GAT_HEADS + h], a);
  }
}

// --------------- aggregation: agg[dst] += alpha * H[src] -------------------
// One wave per edge; lane l covers dims [4l,4l+4) (head = l>>3). float4 gather
// from H[src] + 4 f32 atomic adds into agg[dst] (L2-resident on MI455X).
__global__ __launch_bounds__(256)
void gat_aggregate(const float* __restrict__ eatt, const float* __restrict__ den,
                   const int* __restrict__ src, const int* __restrict__ dst,
                   const float* __restrict__ H, float* __restrict__ agg) {
  const int tid = blockIdx.x * blockDim.x + threadIdx.x;
  const int e = tid >> 5;
  const int l = tid & 31;
  if (e >= GAT_E) return;
  const int s = src[e], d = dst[e];
  const int head = l >> 3;
  const float alpha = eatt[(size_t)e * GAT_HEADS + head] /
                      fmaxf(den[d * GAT_HEADS + head], 1e-9f);
  const float4 hv = *(const float4*)(H + (size_t)s * GAT_F + l * 4);
  float* out = agg + (size_t)d * GAT_F + l * 4;
  atomicAdd(out + 0, alpha * hv.x);
  atomicAdd(out + 1, alpha * hv.y);
  atomicAdd(out + 2, alpha * hv.z);
  atomicAdd(out + 3, alpha * hv.w);
}

// --------------- epilogue: out = elu(agg + b) ------------------------------
__global__ __launch_bounds__(256)
void gat_epilogue(const float* __restrict__ agg, const float* __restrict__ b,
                  float* __restrict__ out, int total) {
  const int i = blockIdx.x * blockDim.x + threadIdx.x;
  if (i >= total) return;
  const float v = agg[i] + b[i & (GAT_F - 1)];
  out[i] = (v > 0.f) ? v : expm1f(v);
}

// ---------------------------------------------------------------------------
static void run_gat_layer(const float* x, const float* W, const float* al,
                          const float* ar, const float* b, const int* src,
                          const int* dst, float* H, float* agg, float* el,
                          float* er, float* mx, float* den, float* ae,
                          float* xout, hipStream_t stream) {
  gat_gemm_wmma<<<GAT_N / 16, 256, 0, stream>>>(x, W, H);
  gat_attn_coef<<<(GAT_N + 255) / 256, 256, 0, stream>>>(H, al, ar, el, er);
  gat_init_nodes<<<(GAT_N * GAT_HEADS + 255) / 256, 256, 0, stream>>>(
      mx, den, GAT_N * GAT_HEADS);
  gat_zero<<<(GAT_N * GAT_F) / 256, 256, 0, stream>>>(agg, GAT_N * GAT_F);
  gat_edge_logits<<<(GAT_E + 255) / 256, 256, 0, stream>>>(el, er, src, dst, ae, mx);
  gat_edge_exp<<<(GAT_E + 255) / 256, 256, 0, stream>>>(ae, dst, mx, den);
  gat_aggregate<<<(GAT_E * 32) / 256, 256, 0, stream>>>(ae, den, src, dst, H, agg);
  gat_epilogue<<<(GAT_N * GAT_F) / 256, 256, 0, stream>>>(agg, b, xout,
                                                          GAT_N * GAT_F);
}

extern "C" void kernel_launch(void* const* d_in, const int* in_sizes, int n_in,
                              void* d_out, int out_size, void* d_ws, size_t ws_size,
                              hipStream_t stream) {
  (void)in_sizes; (void)n_in; (void)out_size; (void)ws_size;
  const float* feats = (const float*)d_in[0];
  const int*   src   = (const int*)d_in[1];
  const int*   dst   = (const int*)d_in[2];
  const float* W1    = (const float*)d_in[3];
  const float* al1   = (const float*)d_in[4];
  const float* ar1   = (const float*)d_in[5];
  const float* b1    = (const float*)d_in[6];
  const float* W2    = (const float*)d_in[7];
  const float* al2   = (const float*)d_in[8];
  const float* ar2   = (const float*)d_in[9];
  const float* b2    = (const float*)d_in[10];

  float* ws   = (float*)d_ws;
  float* buf0 = ws;                                   // H (node features)  25.6MB
  float* buf1 = buf0 + (size_t)GAT_N * GAT_F;         // agg / elu'd x      25.6MB
  float* el   = buf1 + (size_t)GAT_N * GAT_F;         // [N,4]
  float* er   = el + (size_t)GAT_N * GAT_HEADS;
  float* mx   = er + (size_t)GAT_N * GAT_HEADS;
  float* den  = mx + (size_t)GAT_N * GAT_HEADS;
  float* ae   = den + (size_t)GAT_N * GAT_HEADS;      // [E,4]              12.8MB

  // Layer 1: x = feats -> h in buf0, agg+elu in-place in buf1.
  run_gat_layer(feats, W1, al1, ar1, b1, src, dst,
                buf0, buf1, el, er, mx, den, ae, buf1, stream);
  // Layer 2: x = buf1 -> h overwrites buf0; after the GEMM consumes buf1 it is
  // reused as agg (stream-ordered); epilogue writes final ELU straight to d_out.
  run_gat_layer(buf1, W2, al2, ar2, b2, src, dst,
                buf0, buf1, el, er, mx, den, ae, (float*)d_out, stream);
}